// Generator_2327872274973
// MI455X (gfx1250) — compile-verified
//
#include <hip/hip_runtime.h>
#include <hip/hip_bf16.h>

typedef __attribute__((ext_vector_type(2))) float v2f;
typedef __attribute__((ext_vector_type(8))) float v8f;

#define NODES 100000
#define RELS  3
#define EDGES 100000
#define DIN_  32

// ---------------------------------------------------------------- utilities
__global__ __launch_bounds__(256) void k_fill_zero(float* __restrict__ p, long n) {
  long i = (long)blockIdx.x * blockDim.x + threadIdx.x;
  if (i < n) p[i] = 0.0f;
}

__global__ __launch_bounds__(256) void k_deg_count(const int* __restrict__ src,
                                                   const int* __restrict__ dst,
                                                   float* __restrict__ deg_out,
                                                   float* __restrict__ deg_in,
                                                   long re, int n_nodes, int e_per_r) {
  long i = (long)blockIdx.x * blockDim.x + threadIdx.x;
  if (i >= re) return;
  int r = (int)(i / e_per_r);
  atomicAdd(&deg_out[(size_t)r * n_nodes + src[i]], 1.0f);
  atomicAdd(&deg_in [(size_t)r * n_nodes + dst[i]], 1.0f);
}

__global__ __launch_bounds__(256) void k_deg_rsqrt(float* __restrict__ p, long n) {
  long i = (long)blockIdx.x * blockDim.x + threadIdx.x;
  if (i < n) p[i] = rsqrtf(fmaxf(p[i], 1.0f));
}

// ---------------------------------------------------------------- W transpose
// WT[c*K + k] = W[k*Dout + c]; Dout is a power of two (shift/mask indexing).
__global__ __launch_bounds__(256) void k_transpose(const float* __restrict__ W,
                                                   float* __restrict__ WT,
                                                   int K, int d_shift, int d_mask,
                                                   int total) {
  int i = blockIdx.x * blockDim.x + threadIdx.x;
  if (i >= total) return;
  int k = i >> d_shift;
  int c = i & d_mask;
  WT[(size_t)c * K + k] = W[i];
}

// ---------------------------------------------------------------- WMMA GEMM
// Out[N,Dout] = (A[N,K] * rowScale[N]) @ W[K,Dout] (+ bias[Dout]), W given
// TRANSPOSED as WT[Dout,K] so every operand fragment is one contiguous b64
// load. One wave computes a 16x64 strip (4 adjacent 16x16 tiles): each A
// fragment feeds 4 V_WMMA_F32_16X16X4_F32 ops; the 5 loads per K-step go to
// distinct registers so they can overlap the WMMA chain.
// A-operand layout (ISA 7.12.2, 32-bit A 16x4): lanes 0-15 -> K=k0,k0+1;
// lanes 16-31 -> K=k0+2,k0+3; row = lane&15. B mirrored. D: M = vgpr + 8*(lane>=16).
__global__ __launch_bounds__(256) void k_gemm_wmma(const float* __restrict__ A,
                                                   const float* __restrict__ WT,
                                                   const float* __restrict__ rowScale,
                                                   const float* __restrict__ bias,
                                                   float* __restrict__ Out,
                                                   int n_rows, int K, int Dout) {
  const int lane = threadIdx.x & 31;
  const int wave = threadIdx.x >> 5;
  const int colGroups = Dout >> 6;                // 16x64 strips per row band
  const long strip = (long)blockIdx.x * 8 + wave;
  const long totalStrips = (long)(n_rows >> 4) * colGroups;
  if (strip >= totalStrips) return;               // wave-uniform exit (EXEC stays full)
  const int tm = (int)(strip / colGroups);
  const int tg = (int)(strip % colGroups);

  const int rowA  = tm * 16 + (lane & 15);
  const int col0  = tg * 64 + (lane & 15);        // first of 4 tile columns
  const int khalf = (lane >> 4) << 1;             // 0 or 2
  const float s = rowScale ? rowScale[rowA] : 1.0f;

  // Per-lane base pointers (pre-offset by khalf): inner loop uses +k0 only.
  const float* __restrict__ ap  = A  + (size_t)rowA * K + khalf;
  const float* __restrict__ wt0 = WT + (size_t)(col0)      * K + khalf;
  const float* __restrict__ wt1 = WT + (size_t)(col0 + 16) * K + khalf;
  const float* __restrict__ wt2 = WT + (size_t)(col0 + 32) * K + khalf;
  const float* __restrict__ wt3 = WT + (size_t)(col0 + 48) * K + khalf;

  v8f acc0 = {}, acc1 = {}, acc2 = {}, acc3 = {};
#pragma unroll 2
  for (int k0 = 0; k0 < K; k0 += 4) {
    v2f a   = *(const v2f*)(ap  + k0);
    v2f b0v = *(const v2f*)(wt0 + k0);
    v2f b1v = *(const v2f*)(wt1 + k0);
    v2f b2v = *(const v2f*)(wt2 + k0);
    v2f b3v = *(const v2f*)(wt3 + k0);
    a = a * s;                                    // v_pk_mul_f32
    acc0 = __builtin_amdgcn_wmma_f32_16x16x4_f32(false, a, false, b0v, (short)0, acc0, false, false);
    acc1 = __builtin_amdgcn_wmma_f32_16x16x4_f32(false, a, false, b1v, (short)0, acc1, false, false);
    acc2 = __builtin_amdgcn_wmma_f32_16x16x4_f32(false, a, false, b2v, (short)0, acc2, false, false);
    acc3 = __builtin_amdgcn_wmma_f32_16x16x4_f32(false, a, false, b3v, (short)0, acc3, false, false);
  }

  const int rbase = tm * 16 + ((lane >> 4) << 3); // +8 rows for upper half-wave
  float bi0 = 0.f, bi1 = 0.f, bi2 = 0.f, bi3 = 0.f;
  if (bias) {
    bi0 = bias[col0];      bi1 = bias[col0 + 16];
    bi2 = bias[col0 + 32]; bi3 = bias[col0 + 48];
  }
#pragma unroll
  for (int i = 0; i < 8; ++i) {
    float* __restrict__ orow = Out + (size_t)(rbase + i) * Dout + col0;
    orow[0]  = acc0[i] + bi0;
    orow[16] = acc1[i] + bi1;
    orow[32] = acc2[i] + bi2;
    orow[48] = acc3[i] + bi3;
  }
}

// ---------------------------------------------------------------- edge scatter
// acc[dst[e]][:] += proj[src[e]][:] * rsqrt_in[dst[e]]   (float4 per thread)
__global__ __launch_bounds__(256) void k_edge_scatter(const int* __restrict__ src,
                                                      const int* __restrict__ dst,
                                                      const float* __restrict__ rin,
                                                      const float* __restrict__ proj,
                                                      float* __restrict__ acc,
                                                      long total, int d4_shift,
                                                      int d4_mask) {
  long i = (long)blockIdx.x * blockDim.x + threadIdx.x;
  if (i >= total) return;
  int e = (int)(i >> d4_shift);
  int c = (int)(i & d4_mask);
  int s = src[e];
  int d = dst[e];
  float sc = rin[d];
  const int d4 = d4_mask + 1;
  const float4 v = ((const float4*)proj)[(size_t)s * d4 + c];
  float* o = acc + ((size_t)d * d4 + c) * 4;
  atomicAdd(o + 0, v.x * sc);
  atomicAdd(o + 1, v.y * sc);
  atomicAdd(o + 2, v.z * sc);
  atomicAdd(o + 3, v.w * sc);
}

// ---------------------------------------------------------------- finalize
// out[n][c] = acc[n][c] + sum_r B[r][c]   (Dout is a power of two -> mask)
__global__ __launch_bounds__(256) void k_finalize(const float* __restrict__ acc,
                                                  const float* __restrict__ Bst,
                                                  int d_mask, int Dout,
                                                  float* __restrict__ out,
                                                  long total) {
  long i = (long)blockIdx.x * blockDim.x + threadIdx.x;
  if (i >= total) return;
  int c = (int)(i & d_mask);
  float b = 0.0f;
#pragma unroll
  for (int r = 0; r < RELS; ++r) b += Bst[(size_t)r * Dout + c];
  out[i] = acc[i] + b;
}

// ---------------------------------------------------------------- launch
static inline int ilog2(int v) { int s = 0; while ((1 << s) < v) ++s; return s; }

extern "C" void kernel_launch(void* const* d_in, const int* in_sizes, int n_in,
                              void* d_out, int out_size, void* d_ws, size_t ws_size,
                              hipStream_t stream) {
  const float* x  = (const float*)d_in[0];
  const int*   src = (const int*)d_in[1];
  const int*   dst = (const int*)d_in[2];
  const float* W0 = (const float*)d_in[3];
  const float* b0 = (const float*)d_in[4];
  const float* W1 = (const float*)d_in[5];
  const float* b1 = (const float*)d_in[6];
  const float* W2 = (const float*)d_in[7];
  const float* b2 = (const float*)d_in[8];
  const float* W3 = (const float*)d_in[9];
  const float* b3 = (const float*)d_in[10];
  float* out = (float*)d_out;

  float* ws      = (float*)d_ws;
  float* deg_out = ws;                                   // [R*N] -> rsqrt(max(outdeg,1))
  float* deg_in  = deg_out + (size_t)RELS * NODES;       // [R*N] -> rsqrt(max(indeg,1))
  float* H       = deg_in  + (size_t)RELS * NODES;       // [N*256] layer activations
  float* P       = H       + (size_t)NODES * 256;        // [N*256] per-rel projection
  float* ACC     = P       + (size_t)NODES * 256;        // [N*256] scatter accumulator
  float* WT      = ACC     + (size_t)NODES * 256;        // [256*128] transposed weights

  // --- degrees: count once, then rsqrt(max(deg,1)) in place
  {
    long n = 2L * RELS * NODES;
    k_fill_zero<<<(int)((n + 255) / 256), 256, 0, stream>>>(deg_out, n);
    long re = (long)RELS * EDGES;
    k_deg_count<<<(int)((re + 255) / 256), 256, 0, stream>>>(src, dst, deg_out, deg_in,
                                                             re, NODES, EDGES);
    k_deg_rsqrt<<<(int)((n + 255) / 256), 256, 0, stream>>>(deg_out, n);
  }

  auto gemm = [&](const float* A, const float* W, const float* rs, const float* bias,
                  float* O, int K, int Dout) {
    int total = K * Dout;
    int dsh = ilog2(Dout);
    k_transpose<<<(total + 255) / 256, 256, 0, stream>>>(W, WT, K, dsh, Dout - 1, total);
    long strips = (long)(NODES / 16) * (Dout / 64);
    int  blocks = (int)((strips + 7) / 8);               // 8 waves per 256-thread block
    k_gemm_wmma<<<blocks, 256, 0, stream>>>(A, WT, rs, bias, O, NODES, K, Dout);
  };

  auto conv = [&](const float* Hin, int K, int Dout, const float* Wst,
                  const float* Bst, float* Hout) {
    long an = (long)NODES * Dout;
    k_fill_zero<<<(int)((an + 255) / 256), 256, 0, stream>>>(ACC, an);
    int d4 = Dout / 4;
    int sh = ilog2(d4);
    for (int r = 0; r < RELS; ++r) {
      gemm(Hin, Wst + (size_t)r * K * Dout, deg_out + (size_t)r * NODES, nullptr,
           P, K, Dout);
      long ne = (long)EDGES * d4;
      k_edge_scatter<<<(int)((ne + 255) / 256), 256, 0, stream>>>(
          src + (size_t)r * EDGES, dst + (size_t)r * EDGES,
          deg_in + (size_t)r * NODES, P, ACC, ne, sh, d4 - 1);
    }
    k_finalize<<<(int)((an + 255) / 256), 256, 0, stream>>>(ACC, Bst, Dout - 1, Dout,
                                                            Hout, an);
  };

  // layer 0: H = x @ W0 + b0
  gemm(x, W0, nullptr, b0, H, DIN_, 64);
  // conv1..3 (H buffer reused: finalize reads only ACC)
  conv(H, 64, 128, W1, b1, H);
  conv(H, 128, 256, W2, b2, H);
  conv(H, 256, 64, W3, b3, out);
}